// ChebConvModel_17635135718040
// MI455X (gfx1250) — compile-verified
//
#include <hip/hip_runtime.h>

// ChebConv(K=3) x2 + BN + ReLU for MI455X (gfx1250).
// - project-then-propagate: all edge propagation at width 32 (H=30 padded),
//   whole working set L2-resident (192MB).
// - GEMMs: native fp32 WMMA V_WMMA_F32_16X16X4_F32; weight panel staged into
//   LDS via the Tensor Data Mover; panel stored k-pair-interleaved so each B
//   fragment is a single aligned ds_load_b64 (no VGPR repacking).
// - 32x32 output tile per wave (4 accumulators), 8 waves/block -> 256 rows.
// - fused dual-propagation kernel shares edge fetches between the two
//   independent first-hop props.

#define FIN1 128
#define HH   30
#define HP   32
#define BNEPS 1e-5f

typedef __attribute__((ext_vector_type(2))) float        v2f;
typedef __attribute__((ext_vector_type(8))) float        v8f;
typedef __attribute__((ext_vector_type(4))) unsigned int v4u;
typedef __attribute__((ext_vector_type(4))) int          v4i;
typedef __attribute__((ext_vector_type(8))) int          v8i;

__global__ void zero_f32(float* __restrict__ p, long long n) {
  long long i = (long long)blockIdx.x * blockDim.x + threadIdx.x;
  long long stride = (long long)gridDim.x * blockDim.x;
  for (; i < n; i += stride) p[i] = 0.0f;
}

// W1: (3,128,30) -> Wp1: 3 panels of 128x32, k-pair interleaved:
//   Wp[panel][ (f>>1)*64 + c*2 + (f&1) ] = W[k][f][c]  (c>=30 -> 0)
__global__ void pack_w1(const float* __restrict__ W, float* __restrict__ Wp) {
  int t = blockIdx.x * blockDim.x + threadIdx.x;
  if (t >= 3 * FIN1 * HP) return;
  int c = t & (HP - 1);
  int f = (t >> 5) & (FIN1 - 1);
  int k = t >> 12;
  float v = (c < HH) ? W[(k * FIN1 + f) * HH + c] : 0.0f;
  Wp[k * (FIN1 * HP) + (f >> 1) * 64 + c * 2 + (f & 1)] = v;
}

// W2: (3,30,30) -> Wp2: 3 panels of 32x32, k-pair interleaved, zero padded
__global__ void pack_w2(const float* __restrict__ W, float* __restrict__ Wp) {
  int t = blockIdx.x * blockDim.x + threadIdx.x;
  if (t >= 3 * HP * HP) return;
  int c = t & (HP - 1);
  int r = (t >> 5) & (HP - 1);
  int k = t >> 10;
  float v = (r < HH && c < HH) ? W[(k * HH + r) * HH + c] : 0.0f;
  Wp[k * (HP * HP) + (r >> 1) * 64 + c * 2 + (r & 1)] = v;
}

__global__ void degree_k(const int* __restrict__ row, float* __restrict__ deg, int E) {
  int e = blockIdx.x * blockDim.x + threadIdx.x;
  if (e < E) atomicAdd(&deg[row[e]], 1.0f);
}

__global__ void dinv_k(const float* __restrict__ deg, float* __restrict__ dinv, int n) {
  int i = blockIdx.x * blockDim.x + threadIdx.x;
  if (i < n) {
    float d = deg[i];
    dinv[i] = (d > 0.0f) ? rsqrtf(fmaxf(d, 1.0f)) : 0.0f;
  }
}

__global__ void norm_k(const int* __restrict__ row, const int* __restrict__ col,
                       const float* __restrict__ dinv, float* __restrict__ norm, int E) {
  int e = blockIdx.x * blockDim.x + threadIdx.x;
  if (e < E) norm[e] = -dinv[row[e]] * dinv[col[e]];
}

// Stage the fin*32-float weight panel into LDS with the TDM (linear copy, so
// the interleaved layout is preserved). D# per CDNA5 ISA 8.3/8.4.
__device__ __forceinline__ void stage_panel_lds(float* bsm, const float* Wp, int fin) {
#if __has_builtin(__builtin_amdgcn_tensor_load_to_lds) && __has_builtin(__builtin_amdgcn_s_wait_tensorcnt)
  if ((threadIdx.x >> 5) == 0) {                 // wave 0 only (wave-uniform branch)
    unsigned int lds_off = (unsigned int)(size_t)(&bsm[0]);   // low 32b of flat LDS addr = LDS offset
    unsigned long long ga = (unsigned long long)(size_t)Wp;
    v4u g0 = { 1u,                                            // count=1, user mode
               lds_off,                                       // lds_addr[31:0]
               (unsigned int)(ga & 0xFFFFFFFFull),            // global_addr[31:0]
               (unsigned int)((ga >> 32) & 0x1FFFFFFull) | 0x80000000u };  // addr[56:32] | type=2
    v8i g1 = { (int)(2u << 16),                               // data_size=4B
               (int)(32u << 16),                              // tensor_dim0 = 32
               (int)((unsigned int)fin << 16),                // tensor_dim1 = fin
               (int)(32u << 16),                              // tile_dim0 = 32
               fin,                                           // tile_dim1 = fin
               32,                                            // tensor_dim0_stride = 32
               0, 0 };
    v4i gz = { 0, 0, 0, 0 };
#if defined(__clang_major__) && __clang_major__ >= 23
    v8i gz8 = { 0, 0, 0, 0, 0, 0, 0, 0 };
    __builtin_amdgcn_tensor_load_to_lds(g0, g1, gz, gz, gz8, 0);
#else
    __builtin_amdgcn_tensor_load_to_lds(g0, g1, gz, gz, 0);
#endif
    __builtin_amdgcn_s_wait_tensorcnt(0);
  }
  __syncthreads();
#else
  for (int t = threadIdx.x; t < fin * HP; t += 256) bsm[t] = Wp[t];
  __syncthreads();
#endif
}

// Z(N x 32) = X(N x fin) @ W(fin x 32), fp32 WMMA 16x16x4.
// Wave tile = 32 rows x 32 cols: A strips a0 (rows 0-15) / a1 (rows 16-31),
// B fragments b0 (cols 0-15) / b1 (cols 16-31) as single ds_load_b64 each.
__global__ __launch_bounds__(256)
void gemm_wmma(const float* __restrict__ X, const float* __restrict__ Wp,
               float* __restrict__ Z, int nrows, int fin) {
  __shared__ float bsm[FIN1 * HP];               // 16 KB max (of 320 KB/WGP)
  stage_panel_lds(bsm, Wp, fin);

  const int wave = threadIdx.x >> 5;
  const int lane = threadIdx.x & 31;
  const int rowBase = blockIdx.x * 256 + wave * 32;
  if (rowBase >= nrows) return;                  // wave-uniform; EXEC all-1s for WMMA
  const int lo = lane & 15;
  const int hi = lane >> 4;
  int ar0 = rowBase + lo;       if (ar0 >= nrows) ar0 = nrows - 1;
  int ar1 = rowBase + 16 + lo;  if (ar1 >= nrows) ar1 = nrows - 1;

  v8f c00 = {}, c01 = {}, c10 = {}, c11 = {};
  const float* xr0 = X + (size_t)ar0 * fin;
  const float* xr1 = X + (size_t)ar1 * fin;

  for (int k0 = 0; k0 < fin; k0 += 4) {
    const int kr = k0 + 2 * hi;                  // even
    v2f a0 = *(const v2f*)(xr0 + kr);            // 8B aligned
    v2f a1 = *(const v2f*)(xr1 + kr);
    const float* bp = bsm + ((k0 >> 1) + hi) * 64 + 2 * lo;
    v2f b0 = *(const v2f*)(bp);                  // (W[kr][lo],    W[kr+1][lo])
    v2f b1 = *(const v2f*)(bp + 32);             // (W[kr][lo+16], W[kr+1][lo+16])
    c00 = __builtin_amdgcn_wmma_f32_16x16x4_f32(false, a0, false, b0, (short)0, c00, false, false);
    c01 = __builtin_amdgcn_wmma_f32_16x16x4_f32(false, a0, false, b1, (short)0, c01, false, false);
    c10 = __builtin_amdgcn_wmma_f32_16x16x4_f32(false, a1, false, b0, (short)0, c10, false, false);
    c11 = __builtin_amdgcn_wmma_f32_16x16x4_f32(false, a1, false, b1, (short)0, c11, false, false);
  }

  for (int m = 0; m < 8; ++m) {
    int r0 = rowBase + m + 8 * hi;
    int r1 = r0 + 16;
    if (r0 < nrows) {
      Z[(size_t)r0 * HP + lo]      = c00[m];
      Z[(size_t)r0 * HP + 16 + lo] = c01[m];
    }
    if (r1 < nrows) {
      Z[(size_t)r1 * HP + lo]      = c10[m];
      Z[(size_t)r1 * HP + 16 + lo] = c11[m];
    }
  }
}

// Single propagation: out[row[e]][c] += norm[e] * z[col[e]][c]
__global__ void prop_k(const float* __restrict__ z, const int* __restrict__ row,
                       const int* __restrict__ col, const float* __restrict__ norm,
                       float* __restrict__ out, int E) {
  unsigned int t = blockIdx.x * blockDim.x + threadIdx.x;
  int e = t >> 5;
  int c = t & 31;
  if (e < E) {
    float v = norm[e] * z[(size_t)col[e] * HP + c];
    atomicAdd(&out[(size_t)row[e] * HP + c], v);
  }
}

// Fused dual propagation: two independent width-32 props sharing edge fetches
__global__ void prop2_k(const float* __restrict__ za, const float* __restrict__ zb,
                        const int* __restrict__ row, const int* __restrict__ col,
                        const float* __restrict__ norm,
                        float* __restrict__ outa, float* __restrict__ outb, int E) {
  unsigned int t = blockIdx.x * blockDim.x + threadIdx.x;
  int e = t >> 5;
  int c = t & 31;
  if (e < E) {
    float w = norm[e];
    size_t ci = (size_t)col[e] * HP + c;
    size_t ri = (size_t)row[e] * HP + c;
    atomicAdd(&outa[ri], w * za[ci]);
    atomicAdd(&outb[ri], w * zb[ci]);
  }
}

// cheb_out = z0 + p1 + 2*p2b - z2 + b ; relu ; bn -> padded Nx32 buffer
__global__ void epi_pad(const float* __restrict__ z0, const float* __restrict__ z2,
                        const float* __restrict__ p1, const float* __restrict__ p2b,
                        const float* __restrict__ b, const float* __restrict__ gamma,
                        const float* __restrict__ beta, const float* __restrict__ mean,
                        const float* __restrict__ var, float* __restrict__ h, int n) {
  unsigned int t = blockIdx.x * blockDim.x + threadIdx.x;
  int i = t >> 5;
  int c = t & 31;
  if (i >= n) return;
  float v = 0.0f;
  if (c < HH) {
    size_t idx = (size_t)i * HP + c;
    v = z0[idx] + p1[idx] + 2.0f * p2b[idx] - z2[idx] + b[c];
    v = fmaxf(v, 0.0f);
    v = gamma[c] * (v - mean[c]) * rsqrtf(var[c] + BNEPS) + beta[c];
  }
  h[(size_t)i * HP + c] = v;
}

// same epilogue but writes the compact (N,30) output
__global__ void epi_out(const float* __restrict__ z0, const float* __restrict__ z2,
                        const float* __restrict__ p1, const float* __restrict__ p2b,
                        const float* __restrict__ b, const float* __restrict__ gamma,
                        const float* __restrict__ beta, const float* __restrict__ mean,
                        const float* __restrict__ var, float* __restrict__ out, int n) {
  unsigned int t = blockIdx.x * blockDim.x + threadIdx.x;
  int i = t >> 5;
  int c = t & 31;
  if (i >= n || c >= HH) return;
  size_t idx = (size_t)i * HP + c;
  float v = z0[idx] + p1[idx] + 2.0f * p2b[idx] - z2[idx] + b[c];
  v = fmaxf(v, 0.0f);
  v = gamma[c] * (v - mean[c]) * rsqrtf(var[c] + BNEPS) + beta[c];
  out[(size_t)i * HH + c] = v;
}

extern "C" void kernel_launch(void* const* d_in, const int* in_sizes, int n_in,
                              void* d_out, int out_size, void* d_ws, size_t ws_size,
                              hipStream_t stream) {
  const float* x  = (const float*)d_in[0];
  const int*   ei = (const int*)d_in[1];
  // d_in[2] = batch, unused
  const float* W1  = (const float*)d_in[3];
  const float* b1  = (const float*)d_in[4];
  const float* g1  = (const float*)d_in[5];
  const float* be1 = (const float*)d_in[6];
  const float* m1  = (const float*)d_in[7];
  const float* v1  = (const float*)d_in[8];
  const float* W2  = (const float*)d_in[9];
  const float* b2  = (const float*)d_in[10];
  const float* g2  = (const float*)d_in[11];
  const float* be2 = (const float*)d_in[12];
  const float* m2  = (const float*)d_in[13];
  const float* v2  = (const float*)d_in[14];

  const int N = in_sizes[0] / FIN1;
  const int E = in_sizes[1] / 2;
  const int* row = ei;
  const int* col = ei + E;

  // Workspace carve-up (~97 MB for N=100k, E=1.6M)
  float* ws   = (float*)d_ws;
  float* deg  = ws; ws += N;
  float* dinv = ws; ws += N;
  float* norm = ws; ws += E;
  float* Wp1  = ws; ws += 3 * FIN1 * HP;
  float* Wp2  = ws; ws += 3 * HP * HP;
  const size_t NP = (size_t)N * HP;
  float* z0  = ws; ws += NP;
  float* z1  = ws; ws += NP;
  float* z2  = ws; ws += NP;
  float* p1  = ws; ws += NP;   // p1, p2, p2b contiguous -> single zero-fill
  float* p2  = ws; ws += NP;
  float* p2b = ws; ws += NP;
  float* hb  = ws; ws += NP;

  const int eg  = (E + 255) / 256;
  const int ng  = (N + 255) / 256;
  const int pg  = (int)(((long long)E * 32 + 255) / 256);
  const int epg = (int)(((long long)N * 32 + 255) / 256);
  const int gg  = (N + 255) / 256;               // 256 rows per block now

  // ---- graph prep ----
  zero_f32<<<1024, 256, 0, stream>>>(deg, N);
  degree_k<<<eg, 256, 0, stream>>>(row, deg, E);
  dinv_k<<<ng, 256, 0, stream>>>(deg, dinv, N);
  norm_k<<<eg, 256, 0, stream>>>(row, col, dinv, norm, E);
  pack_w1<<<(3 * FIN1 * HP + 255) / 256, 256, 0, stream>>>(W1, Wp1);
  pack_w2<<<(3 * HP * HP + 255) / 256, 256, 0, stream>>>(W2, Wp2);

  // ---- layer 1: z_k = x @ W1[k], then propagate at width 32 ----
  gemm_wmma<<<gg, 256, 0, stream>>>(x, Wp1,                 z0, N, FIN1);
  gemm_wmma<<<gg, 256, 0, stream>>>(x, Wp1 + 1 * FIN1 * HP, z1, N, FIN1);
  gemm_wmma<<<gg, 256, 0, stream>>>(x, Wp1 + 2 * FIN1 * HP, z2, N, FIN1);

  zero_f32<<<4096, 256, 0, stream>>>(p1, (long long)(3 * NP));
  prop2_k<<<pg, 256, 0, stream>>>(z1, z2, row, col, norm, p1, p2, E);
  prop_k<<<pg, 256, 0, stream>>>(p2, row, col, norm, p2b, E);

  epi_pad<<<epg, 256, 0, stream>>>(z0, z2, p1, p2b, b1, g1, be1, m1, v1, hb, N);

  // ---- layer 2: y_k = h @ W2[k], propagate, epilogue to d_out ----
  gemm_wmma<<<gg, 256, 0, stream>>>(hb, Wp2,               z0, N, HP);
  gemm_wmma<<<gg, 256, 0, stream>>>(hb, Wp2 + 1 * HP * HP, z1, N, HP);
  gemm_wmma<<<gg, 256, 0, stream>>>(hb, Wp2 + 2 * HP * HP, z2, N, HP);

  zero_f32<<<4096, 256, 0, stream>>>(p1, (long long)(3 * NP));
  prop2_k<<<pg, 256, 0, stream>>>(z1, z2, row, col, norm, p1, p2, E);
  prop_k<<<pg, 256, 0, stream>>>(p2, row, col, norm, p2b, E);

  epi_out<<<epg, 256, 0, stream>>>(z0, z2, p1, p2b, b2, g2, be2, m2, v2, (float*)d_out, N);
}